// CentroidDistance_54915451847202
// MI455X (gfx1250) — compile-verified
//
#include <hip/hip_runtime.h>
#include <hip/hip_bf16.h>
#include <math.h>

typedef float v2f __attribute__((ext_vector_type(2)));
typedef float v8f __attribute__((ext_vector_type(8)));

typedef __attribute__((address_space(1))) int gas_int;   // global
typedef __attribute__((address_space(3))) int las_int;   // LDS

#define EMBED     128
#define NCENT     64
#define BLK_ROWS  128          // nodes per workgroup (8 waves x 16 rows)
#define THREADS   256
#define BPAD      (NCENT + 1)  // B row stride in float2 units (padding vs bank conflicts)
#define EPS_F     1e-5f

#if __has_builtin(__builtin_amdgcn_global_load_async_to_lds_b32)
#define HAVE_ASYNC_LDS 1
#else
#define HAVE_ASYNC_LDS 0
#endif

// ---------------------------------------------------------------------------
// One block: 128 nodes x 64 centroids via V_WMMA_F32_16X16X4_F32.
// B staged in LDS in WMMA-fragment order: row (kstep,half) holds float2
// pairs (B[k][c], B[k+1][c]) so each fragment is ONE ds_load_b64 into an
// even-aligned VGPR pair feeding WMMA SRC1 directly (no register shuffling).
// ---------------------------------------------------------------------------
__global__ __launch_bounds__(THREADS) void cd_main(
    const float* __restrict__ node_repr,   // [N, 128]
    const float* __restrict__ mask,        // [N, 1]
    const float* __restrict__ cw,          // [64, 128]
    float* __restrict__ out_node,          // [N, 64]
    float* __restrict__ out_graph,         // [64] (pre-zeroed; atomic accumulate)
    int nnodes)
{
    __shared__ float2 Bsh2[(EMBED / 2) * BPAD]; // 64 rows x (64+1) float2 = 33.3 KB
    __shared__ float  sv_sh[NCENT];             // ||v_c||^2
    __shared__ float  msk_sh[BLK_ROWS];         // per-row mask (0 for OOB rows)
    __shared__ float  colsum[NCENT];            // per-centroid partial sums

    const int tid  = threadIdx.x;
    const int wave = tid >> 5;
    const int lane = tid & 31;
    const int m    = lane & 15;   // row (A) / col-in-tile (B,C)
    const int half = lane >> 4;   // 0|1: selects K pair (A,B) / row offset +8 (C)
    const int rowbase = blockIdx.x * BLK_ROWS;

    // ---- stage B = cw^T, pair-packed per (kstep,half) row -------------------
    // element e of centroid c (k = e): row = (e>>2)*2 + ((e>>1)&1), sub = e&1
    // float index = row*(2*BPAD) + 2*c + sub
#if HAVE_ASYNC_LDS
    {
        float* Bf = (float*)Bsh2;
        for (int i = tid; i < NCENT * EMBED; i += THREADS) {
            int c = i >> 7;
            int e = i & (EMBED - 1);
            int row  = ((e >> 2) << 1) | ((e >> 1) & 1);
            int fidx = row * (2 * BPAD) + (c << 1) + (e & 1);
            __builtin_amdgcn_global_load_async_to_lds_b32(
                (gas_int*)(cw + i),      // global source (ASYNCcnt-tracked)
                (las_int*)(Bf + fidx),   // LDS destination (permuted pair-pack)
                0, 0);
        }
        asm volatile("s_wait_asynccnt 0x0" ::: "memory");
    }
#else
    {
        float* Bf = (float*)Bsh2;
        for (int i = tid; i < NCENT * EMBED; i += THREADS) {
            int c = i >> 7;
            int e = i & (EMBED - 1);
            int row  = ((e >> 2) << 1) | ((e >> 1) & 1);
            Bf[row * (2 * BPAD) + (c << 1) + (e & 1)] = cw[i];
        }
    }
#endif
    // ---- per-row mask & per-centroid norms ---------------------------------
    if (tid < BLK_ROWS) {
        int gn = rowbase + tid;
        msk_sh[tid] = (gn < nnodes) ? mask[gn] : 0.0f;
    } else if (tid < BLK_ROWS + NCENT) {
        int c = tid - BLK_ROWS;
        const float4* p = (const float4*)(cw + c * EMBED);
        float s = 0.0f;
        #pragma unroll
        for (int j = 0; j < EMBED / 4; ++j) {
            float4 v = p[j];
            s += v.x * v.x + v.y * v.y + v.z * v.z + v.w * v.w;
        }
        sv_sh[c] = s;
    }
    if (tid < NCENT) colsum[tid] = 0.0f;
    __syncthreads();

    // ---- f32 WMMA GEMM: each wave -> 16 rows x 64 cols, K=128 step 4 -------
    // A 16x4 layout: lane m holds row m; VGPR0/1 = K = 2*half + {0,1}
    // B 4x16 layout: lane m holds col m; VGPR0/1 = K = 2*half + {0,1}
    const int node = rowbase + wave * 16 + m;
    const int ncl  = (node < nnodes) ? node : (nnodes - 1);  // clamp: EXEC stays all-1s
    const float* arow = node_repr + (size_t)ncl * EMBED + half * 2;

    v8f acc0 = {}, acc1 = {}, acc2 = {}, acc3 = {};
    float sunorm = 0.0f;  // this lane's half of ||u_row||^2 (free from A stream)

    #pragma unroll 4
    for (int kk = 0; kk < EMBED; kk += 4) {
        v2f a = *(const v2f*)(arow + kk);
        sunorm += a.x * a.x + a.y * a.y;
        const float2* brow = &Bsh2[(((kk >> 2) << 1) | half) * BPAD];
        v2f b0 = *(const v2f*)(brow + m +  0);   // ds_load_b64 -> aligned pair
        v2f b1 = *(const v2f*)(brow + m + 16);
        v2f b2 = *(const v2f*)(brow + m + 32);
        v2f b3 = *(const v2f*)(brow + m + 48);
        acc0 = __builtin_amdgcn_wmma_f32_16x16x4_f32(false, a, false, b0, (short)0, acc0, false, false);
        acc1 = __builtin_amdgcn_wmma_f32_16x16x4_f32(false, a, false, b1, (short)0, acc1, false, false);
        acc2 = __builtin_amdgcn_wmma_f32_16x16x4_f32(false, a, false, b2, (short)0, acc2, false, false);
        acc3 = __builtin_amdgcn_wmma_f32_16x16x4_f32(false, a, false, b3, (short)0, acc3, false, false);
    }

    // combine the two K-halves of ||u||^2 (lane m & lane m+16 hold row m)
    float sufull = sunorm + __shfl_xor(sunorm, 16, 32);

    // C/D layout: VGPR r, lane -> row M = r + 8*half, col N = m
    const int rbase = wave * 16 + half * 8;
    float su_r[8], mk_r[8];
    #pragma unroll
    for (int r = 0; r < 8; ++r) {
        su_r[r] = __shfl(sufull, half * 8 + r, 32);  // su of wave-row (8*half + r)
        mk_r[r] = msk_sh[rbase + r];                 // 0 for OOB tail rows
    }

    v8f accs[4] = { acc0, acc1, acc2, acc3 };
    #pragma unroll
    for (int t = 0; t < 4; ++t) {
        const int col = t * 16 + m;
        const float sv = sv_sh[col];
        const float onemsv = 1.0f - sv;
        float csum = 0.0f;
        #pragma unroll
        for (int r = 0; r < 8; ++r) {
            float dot = accs[t][r];
            float sq  = fmaxf(su_r[r] + sv - 2.0f * dot, 0.0f);
            float den = fmaxf((1.0f - su_r[r]) * onemsv, EPS_F);
            float arg = fmaxf(1.0f + 2.0f * sq / den, 1.0f + EPS_F);
            float dist = acoshf(arg) * mk_r[r];
            int gn = rowbase + rbase + r;
            if (gn < nnodes) out_node[(size_t)gn * NCENT + col] = dist;
            csum += dist;  // 0 for OOB rows (mask==0)
        }
        atomicAdd(&colsum[col], csum);  // ds_add_f32
    }

    __syncthreads();
    if (tid < NCENT) atomicAdd(&out_graph[tid], colsum[tid]);  // global_atomic_add_f32
}

// ---------------------------------------------------------------------------
__global__ void cd_init(float* __restrict__ out_graph) {
    if (threadIdx.x < NCENT) out_graph[threadIdx.x] = 0.0f;
}

__global__ __launch_bounds__(256) void cd_final(
    const float* __restrict__ mask, float* __restrict__ out_graph, int nnodes)
{
    __shared__ float red[256];
    float s = 0.0f;
    for (int i = threadIdx.x; i < nnodes; i += 256) s += mask[i];
    red[threadIdx.x] = s;
    __syncthreads();
    for (int off = 128; off > 0; off >>= 1) {
        if (threadIdx.x < (unsigned)off) red[threadIdx.x] += red[threadIdx.x + off];
        __syncthreads();
    }
    if (threadIdx.x < NCENT) out_graph[threadIdx.x] /= red[0];
}

// ---------------------------------------------------------------------------
extern "C" void kernel_launch(void* const* d_in, const int* in_sizes, int n_in,
                              void* d_out, int out_size, void* d_ws, size_t ws_size,
                              hipStream_t stream) {
    const float* node_repr = (const float*)d_in[0];
    const float* mask      = (const float*)d_in[1];
    const float* cw        = (const float*)d_in[2];
    float* out       = (float*)d_out;
    float* out_graph = out;          // [64]
    float* out_node  = out + NCENT;  // [N, 64]

    const int nnodes = in_sizes[0] / EMBED;
    const int nblk   = (nnodes + BLK_ROWS - 1) / BLK_ROWS;

    cd_init<<<1, 64, 0, stream>>>(out_graph);
    cd_main<<<nblk, THREADS, 0, stream>>>(node_repr, mask, cw, out_node, out_graph, nnodes);
    cd_final<<<1, 256, 0, stream>>>(mask, out_graph, nnodes);
}